// NoisyTopkRouter_42949672961981
// MI455X (gfx1250) — compile-verified
//
#include <hip/hip_runtime.h>
#include <math.h>

typedef float v2f __attribute__((ext_vector_type(2)));
typedef float v8f __attribute__((ext_vector_type(8)));

#define N_EMBD 192
#define NE 6
#define WAVES_PER_BLOCK 8

// One wave32 handles a tile of 16 tokens.
// WMMA: A = tokens[16 x 192] (f32), B = [192 x 16] with cols 0-5 = W_route^T,
// cols 6-11 = W_noise^T, cols 12-15 = garbage (independent columns, never read).
// Accumulated via 48 chained V_WMMA_F32_16X16X4_F32.
__global__ __launch_bounds__(256) void noisy_topk_router_kernel(
    const float* __restrict__ mh,     // [T,192]
    const float* __restrict__ noise,  // [T,6]
    const float* __restrict__ Wr,     // [6,192]
    const float* __restrict__ br,     // [6]
    const float* __restrict__ Wn,     // [6,192]
    const float* __restrict__ bn,     // [6]
    const float* __restrict__ tptr,   // [1]
    float* __restrict__ out,          // [T*6] probs ++ [T*2] indices (as float)
    int T)
{
    // per-wave 16x6 noisy-logit tile, row stride 9 words -> conflict-free banks
    __shared__ float lds_noisy[WAVES_PER_BLOCK][16][9];

    const int lane = threadIdx.x & 31;
    const int wid  = threadIdx.x >> 5;
    const int n    = lane & 15;   // A: row (token-in-tile); B/D: column (expert slot)
    const int half = lane >> 4;
    const int kofs = half * 2;    // 32-bit A/B lane-half K offset (ISA 7.12.2)
    const int base = (blockIdx.x * WAVES_PER_BLOCK + wid) * 16;
    const bool active = (base < T);

    v8f acc = {};

    if (active) {
        // A: lane reads row (base+n), K = k0+kofs .. +1 (8B aligned, streamed once)
        const float* aptr = mh + (size_t)(base + n) * N_EMBD + kofs;
        // B: column n -> W_route row n (n<6) / W_noise row n-6 (6<=n<12);
        // n>=12 columns are never consumed, point them at a valid dummy row.
        const float* wrow = (n < NE) ? (Wr + n * N_EMBD)
                                     : (Wn + ((n - NE) % NE) * N_EMBD);
        wrow += kofs;

        #pragma unroll 4
        for (int k0 = 0; k0 < N_EMBD; k0 += 4) {
            v2f a = __builtin_nontemporal_load((const v2f*)(aptr + k0));
            v2f b = *(const v2f*)(wrow + k0);
            // (neg_a, A, neg_b, B, c_mod, C, reuse_a, reuse_b)
            acc = __builtin_amdgcn_wmma_f32_16x16x4_f32(
                false, a, false, b, (short)0, acc, false, false);
        }

        // D layout: VGPR v holds token m = v + half*8 at column n.
        float bias = 0.f;
        if (n < NE)            bias = br[n];
        else if (n < 2 * NE)   bias = bn[n - NE];
        const float temp = fminf(fmaxf(tptr[0], 0.5f), 2.0f);

        #pragma unroll
        for (int v = 0; v < 8; ++v) {
            const int m = v + half * 8;
            const float val = acc[v] + bias;              // logit or noise-logit
            float contrib = 0.f;
            if (n >= NE && n < 2 * NE) {
                const float nz = noise[(size_t)(base + m) * NE + (n - NE)];
                // stable softplus: max(x,0) + log1p(exp(-|x|))
                const float sp = fmaxf(val, 0.f) + log1pf(expf(-fabsf(val)));
                contrib = temp * nz * sp;
            }
            // route lane e pulls noise contribution from lane e+6 (same half)
            const int src = (n < NE) ? ((half << 4) | (n + NE)) : lane;
            const float other = __shfl(contrib, src, 32);
            if (n < NE) lds_noisy[wid][m][n] = val + other;
        }
    }

    __syncthreads();  // order LDS stores (cols-on-lanes) vs loads (rows-on-lanes)

    if (active && lane < 16) {
        const int token = base + lane;
        // top-2 over 6 experts; strict '>' keeps lowest index on ties (matches top_k)
        float v0 = -INFINITY, v1 = -INFINITY;
        int   i0 = 0, i1 = 0;
        #pragma unroll
        for (int e = 0; e < NE; ++e) {
            const float x = lds_noisy[wid][lane][e];
            if (x > v0)      { v1 = v0; i1 = i0; v0 = x; i0 = e; }
            else if (x > v1) { v1 = x;  i1 = e; }
        }
        // softmax over {v0, v1}, others masked to -inf -> prob 0
        const float s   = expf(v1 - v0);
        const float inv = 1.0f / (1.0f + s);
        const float p0 = inv, p1 = s * inv;

        float* ro = out + (size_t)token * NE;
        #pragma unroll
        for (int e = 0; e < NE; ++e)
            ro[e] = (e == i0) ? p0 : ((e == i1) ? p1 : 0.f);

        float* oi = out + (size_t)T * NE + (size_t)token * 2;
        oi[0] = (float)i0;
        oi[1] = (float)i1;
    }
}

extern "C" void kernel_launch(void* const* d_in, const int* in_sizes, int n_in,
                              void* d_out, int out_size, void* d_ws, size_t ws_size,
                              hipStream_t stream) {
    const float* mh    = (const float*)d_in[0];
    const float* noise = (const float*)d_in[1];
    const float* Wr    = (const float*)d_in[2];
    const float* br    = (const float*)d_in[3];
    const float* Wn    = (const float*)d_in[4];
    const float* bn    = (const float*)d_in[5];
    const float* temp  = (const float*)d_in[6];
    float* out = (float*)d_out;

    const int T     = in_sizes[0] / N_EMBD;          // 131072 tokens
    const int tiles = (T + 15) / 16;                  // one wave per tile
    const int blocks = (tiles + WAVES_PER_BLOCK - 1) / WAVES_PER_BLOCK;

    noisy_topk_router_kernel<<<blocks, WAVES_PER_BLOCK * 32, 0, stream>>>(
        mh, noise, Wr, br, Wn, bn, temp, out, T);
}